// Asymm_3d_spconv_27178553049606
// MI455X (gfx1250) — compile-verified
//
#include <hip/hip_runtime.h>

typedef __attribute__((ext_vector_type(2))) float v2f;
typedef __attribute__((ext_vector_type(8))) float v8f;
typedef __attribute__((ext_vector_type(4))) unsigned int u32x4;
typedef __attribute__((ext_vector_type(8))) int i32x8;
typedef __attribute__((ext_vector_type(4))) int i32x4;

#define C_OUT 32
#define SLOPE 0.01f
#define BN_EPS 1e-5f

struct Offs9 { int o[9]; };

enum { MODE_LRELU_BN = 0, MODE_LRELU_BN_ADD = 1 };

// ---------------------------------------------------------------------------
// TDM weight staging: DMA a contiguous fp32 block (E elements) from global
// memory into LDS via the gfx1250 Tensor Data Mover. D# built per
// CDNA5 ISA: group0 = {count=1 | lds_addr | global_addr |
// type=2}, group1 = {data_size=4B, 1-row tile of E elements}.
// This toolchain exposes the 6-arg builtin (g0, g1, g2, g3, g4, cpol).
// Caller must be a single full wave; completes with s_wait_tensorcnt 0.
// ---------------------------------------------------------------------------
#if __has_builtin(__builtin_amdgcn_tensor_load_to_lds) && __has_builtin(__builtin_amdgcn_s_wait_tensorcnt)
#define HAVE_TDM 1
__device__ __forceinline__ void tdm_stage_f32(const float* gsrc, unsigned lds_byte_addr,
                                              unsigned nelem)
{
    const unsigned long long ga = (unsigned long long)(uintptr_t)gsrc;
    u32x4 g0;
    g0.x = 1u;                                            // count=1, no gather, user mode
    g0.y = lds_byte_addr;                                 // LDS dest (bytes)
    g0.z = (unsigned int)(ga & 0xffffffffu);              // global_addr[31:0]
    g0.w = (unsigned int)((ga >> 32) & 0x01ffffffu)       // global_addr[56:32]
         | (2u << 30);                                    // type=2 ("image")
    i32x8 g1;
    g1[0] = (int)(2u << 16);                              // data_size=2 (4 bytes)
    g1[1] = (int)((nelem & 0xffffu) << 16);               // tensor_dim0[15:0]
    g1[2] = (int)((nelem >> 16) | (1u << 16));            // tensor_dim0[31:16] | tensor_dim1=1
    g1[3] = (int)((nelem & 0xffffu) << 16);               // tensor_dim1_hi=0 | tile_dim0=nelem
    g1[4] = 1;                                            // tile_dim1=1, tile_dim2=0
    g1[5] = (int)nelem;                                   // tensor_dim0_stride lo
    g1[6] = 0;
    g1[7] = 0;
    i32x4 g2 = {0, 0, 0, 0};
    i32x4 g3 = {0, 0, 0, 0};
    i32x8 g4 = {0, 0, 0, 0, 0, 0, 0, 0};
    __builtin_amdgcn_tensor_load_to_lds(g0, g1, g2, g3, g4, 0);
}
#else
#define HAVE_TDM 0
#endif

// ---------------------------------------------------------------------------
// Submanifold sparse conv via fp32 WMMA (V_WMMA_F32_16X16X4_F32).
// One wave32 = one tile of 16 voxels; D = 16 voxels x 32 couts (two 16x16 f32
// accumulators sharing each gathered A fragment).
// ---------------------------------------------------------------------------
template <int CIN, int KOFF>
__global__ __launch_bounds__(256)
void subm_conv_wmma(const float* __restrict__ feat,   // [n, CIN]
                    const int*   __restrict__ nbr,    // [27, n]
                    const float* __restrict__ W,      // [KOFF, CIN, 32]
                    const float* __restrict__ bnp,    // [4, 32]
                    const float* __restrict__ addsrc, // optional residual [n, 32]
                    float*       __restrict__ out,    // [n, 32]
                    int n, Offs9 offs, int mode)
{
    __shared__ float lw[KOFF * CIN * C_OUT];
    __shared__ float lbn[4 * C_OUT];
#if HAVE_TDM
    if (threadIdx.x < 32) {                                  // wave 0 issues the DMA
        tdm_stage_f32(W, (unsigned)(uintptr_t)(void*)lw, KOFF * CIN * C_OUT);
        __builtin_amdgcn_s_wait_tensorcnt(0);
    }
#else
    for (int i = threadIdx.x; i < KOFF * CIN * C_OUT; i += blockDim.x) lw[i] = W[i];
#endif
    for (int i = threadIdx.x; i < 4 * C_OUT; i += blockDim.x) lbn[i] = bnp[i];
    __syncthreads();

    const int lane   = threadIdx.x & 31;
    const int mrow   = lane & 15;   // A: voxel row; B/D: cout column (within 16)
    const int khalf  = lane >> 4;   // selects K {0,1} vs {2,3} of each 4-chunk
    const int wavesPerBlock = blockDim.x >> 5;
    const int wave   = blockIdx.x * wavesPerBlock + (threadIdx.x >> 5);
    const int nwaves = gridDim.x * wavesPerBlock;
    const int ntiles = (n + 15) >> 4;

    for (int tile = wave; tile < ntiles; tile += nwaves) {
        const int base = tile << 4;
        const int vrow = base + mrow;
        const bool full = (base + 16) <= n;                  // wave-uniform

        if (tile + nwaves < ntiles) {                        // global_prefetch_b8
            __builtin_prefetch(&nbr[(size_t)offs.o[0] * n + (((tile + nwaves) << 4) + mrow)], 0, 0);
        }

        int gidx[KOFF];
        if (full) {
#pragma unroll
            for (int k = 0; k < KOFF; ++k) gidx[k] = nbr[(size_t)offs.o[k] * n + vrow];
        } else {
#pragma unroll
            for (int k = 0; k < KOFF; ++k)
                gidx[k] = (vrow < n) ? nbr[(size_t)offs.o[k] * n + vrow] : -1;
        }

        v8f accs[2];
        accs[0] = (v8f){0.f, 0.f, 0.f, 0.f, 0.f, 0.f, 0.f, 0.f};
        accs[1] = (v8f){0.f, 0.f, 0.f, 0.f, 0.f, 0.f, 0.f, 0.f};

#pragma unroll
        for (int k = 0; k < KOFF; ++k) {
            const int id = gidx[k];
            const float* src = feat + (size_t)(id < 0 ? 0 : id) * CIN;  // clamped, always valid
            const float msk = (id < 0) ? 0.f : 1.f;
#pragma unroll
            for (int kb = 0; kb < CIN / 4; ++kb) {
                const int kc = kb * 4 + 2 * khalf;
                v2f a = *(const v2f*)(src + kc);
                a.x *= msk; a.y *= msk;                      // EXEC stays all-ones for WMMA
                v2f b0, b1;
                b0.x = lw[(k * CIN + kc    ) * C_OUT + mrow];
                b0.y = lw[(k * CIN + kc + 1) * C_OUT + mrow];
                b1.x = lw[(k * CIN + kc    ) * C_OUT + 16 + mrow];
                b1.y = lw[(k * CIN + kc + 1) * C_OUT + 16 + mrow];
                accs[0] = __builtin_amdgcn_wmma_f32_16x16x4_f32(
                    false, a, false, b0, (short)0, accs[0], false, false);
                accs[1] = __builtin_amdgcn_wmma_f32_16x16x4_f32(
                    false, a, false, b1, (short)0, accs[1], false, false);
            }
        }

        // Epilogue: D element [M = r + 8*khalf][N = lane&15]
#pragma unroll
        for (int cc = 0; cc < 2; ++cc) {
            const int ch = cc * 16 + mrow;
            const float gamma = lbn[ch], beta = lbn[32 + ch];
            const float mean  = lbn[64 + ch], var = lbn[96 + ch];
            const float scale = gamma * rsqrtf(var + BN_EPS);
            const float shift = beta - mean * scale;
            const size_t rowoff = (size_t)(base + 8 * khalf) * C_OUT + ch;
            float* op = out + rowoff;
            const float* ap = addsrc ? (addsrc + rowoff) : nullptr;
            if (full) {
#pragma unroll
                for (int r = 0; r < 8; ++r) {
                    float f = accs[cc][r];
                    f = (f >= 0.f) ? f : SLOPE * f;
                    float v = f * scale + shift;
                    if (mode == MODE_LRELU_BN_ADD) v += ap[r * C_OUT];
                    op[r * C_OUT] = v;
                }
            } else {
#pragma unroll
                for (int r = 0; r < 8; ++r) {
                    const int voxel = base + r + 8 * khalf;
                    if (voxel < n) {
                        float f = accs[cc][r];
                        f = (f >= 0.f) ? f : SLOPE * f;
                        float v = f * scale + shift;
                        if (mode == MODE_LRELU_BN_ADD) v += ap[r * C_OUT];
                        op[r * C_OUT] = v;
                    }
                }
            }
        }
    }
}

// ---------------------------------------------------------------------------
// Fused ReconBlock: three axial 3-tap convs (32->32) over the SAME input y,
// sharing the center tap (7 unique gathers instead of 9), each conv through
// its own BN+sigmoid, summed, then the gating multiply with y -> out.
// ---------------------------------------------------------------------------
__global__ __launch_bounds__(256)
void recon_gate_wmma(const float* __restrict__ y,     // [n, 32]
                     const int*   __restrict__ nbr,   // [27, n]
                     const float* __restrict__ Wr1,   // [3, 32, 32]
                     const float* __restrict__ Wr2,
                     const float* __restrict__ Wr3,
                     const float* __restrict__ bnp,   // [3][4][32] (bn sets 4..6)
                     float*       __restrict__ out,   // [n, 32]
                     int n)
{
    constexpr int CIN = 32;
    __shared__ float lw[3 * 3 * CIN * C_OUT];          // [conv][tap][cin][cout]
    __shared__ float lbn[3 * 4 * C_OUT];
#if HAVE_TDM
    if (threadIdx.x < 32) {
        const unsigned lbase = (unsigned)(uintptr_t)(void*)lw;
        const unsigned E = 3 * CIN * C_OUT;            // 3072 floats per conv
        tdm_stage_f32(Wr1, lbase + 0 * E * 4, E);
        tdm_stage_f32(Wr2, lbase + 1 * E * 4, E);
        tdm_stage_f32(Wr3, lbase + 2 * E * 4, E);
        __builtin_amdgcn_s_wait_tensorcnt(0);
    }
#else
    for (int i = threadIdx.x; i < 3 * CIN * C_OUT; i += blockDim.x) {
        lw[0 * 3 * CIN * C_OUT + i] = Wr1[i];
        lw[1 * 3 * CIN * C_OUT + i] = Wr2[i];
        lw[2 * 3 * CIN * C_OUT + i] = Wr3[i];
    }
#endif
    for (int i = threadIdx.x; i < 3 * 4 * C_OUT; i += blockDim.x) lbn[i] = bnp[i];
    __syncthreads();

    // Unique taps of P311={4,13,22}, P131={10,13,16}, P113={12,13,14}
    constexpr int uoff[7] = {4, 10, 12, 13, 14, 16, 22};
    // tap_of[conv][u] = tap index within that conv's weight tensor, or -1
    constexpr int tap_of[3][7] = {
        { 0, -1, -1,  1, -1, -1,  2},   // P311
        {-1,  0, -1,  1, -1,  2, -1},   // P131
        {-1, -1,  0,  1,  2, -1, -1}};  // P113

    const int lane   = threadIdx.x & 31;
    const int mrow   = lane & 15;
    const int khalf  = lane >> 4;
    const int wavesPerBlock = blockDim.x >> 5;
    const int wave   = blockIdx.x * wavesPerBlock + (threadIdx.x >> 5);
    const int nwaves = gridDim.x * wavesPerBlock;
    const int ntiles = (n + 15) >> 4;

    for (int tile = wave; tile < ntiles; tile += nwaves) {
        const int base = tile << 4;
        const int vrow = base + mrow;
        const bool full = (base + 16) <= n;

        if (tile + nwaves < ntiles) {
            __builtin_prefetch(&nbr[(size_t)13 * n + (((tile + nwaves) << 4) + mrow)], 0, 0);
        }

        int gidx[7];
        if (full) {
#pragma unroll
            for (int u = 0; u < 7; ++u) gidx[u] = nbr[(size_t)uoff[u] * n + vrow];
        } else {
#pragma unroll
            for (int u = 0; u < 7; ++u)
                gidx[u] = (vrow < n) ? nbr[(size_t)uoff[u] * n + vrow] : -1;
        }

        v8f acc[3][2];
#pragma unroll
        for (int c = 0; c < 3; ++c)
#pragma unroll
            for (int cc = 0; cc < 2; ++cc)
                acc[c][cc] = (v8f){0.f, 0.f, 0.f, 0.f, 0.f, 0.f, 0.f, 0.f};

#pragma unroll
        for (int u = 0; u < 7; ++u) {
            const int id = gidx[u];
            const float* src = y + (size_t)(id < 0 ? 0 : id) * CIN;
            const float msk = (id < 0) ? 0.f : 1.f;
#pragma unroll
            for (int kb = 0; kb < CIN / 4; ++kb) {
                const int kc = kb * 4 + 2 * khalf;
                v2f a = *(const v2f*)(src + kc);
                a.x *= msk; a.y *= msk;
#pragma unroll
                for (int c = 0; c < 3; ++c) {
                    const int t = tap_of[c][u];
                    if (t >= 0) {                         // compile-time folded
                        const int wb = ((c * 3 + t) * CIN + kc) * C_OUT;
                        v2f b0, b1;
                        b0.x = lw[wb + mrow];
                        b0.y = lw[wb + C_OUT + mrow];
                        b1.x = lw[wb + 16 + mrow];
                        b1.y = lw[wb + C_OUT + 16 + mrow];
                        acc[c][0] = __builtin_amdgcn_wmma_f32_16x16x4_f32(
                            false, a, false, b0, (short)0, acc[c][0], false, false);
                        acc[c][1] = __builtin_amdgcn_wmma_f32_16x16x4_f32(
                            false, a, false, b1, (short)0, acc[c][1], false, false);
                    }
                }
            }
        }

        // Epilogue: gate = sum_c sigmoid(bn_c(conv_c)); out = gate * y
#pragma unroll
        for (int cc = 0; cc < 2; ++cc) {
            const int ch = cc * 16 + mrow;
            float scale[3], shift[3];
#pragma unroll
            for (int c = 0; c < 3; ++c) {
                const float gamma = lbn[c * 128 + ch], beta = lbn[c * 128 + 32 + ch];
                const float mean  = lbn[c * 128 + 64 + ch], var = lbn[c * 128 + 96 + ch];
                scale[c] = gamma * rsqrtf(var + BN_EPS);
                shift[c] = beta - mean * scale[c];
            }
            const size_t rowoff = (size_t)(base + 8 * khalf) * C_OUT + ch;
            const float* yp = y + rowoff;
            float* op = out + rowoff;
            if (full) {
#pragma unroll
                for (int r = 0; r < 8; ++r) {
                    float g = 0.f;
#pragma unroll
                    for (int c = 0; c < 3; ++c) {
                        const float bnv = acc[c][cc][r] * scale[c] + shift[c];
                        g += 1.f / (1.f + __expf(-bnv));
                    }
                    op[r * C_OUT] = g * yp[r * C_OUT];
                }
            } else {
#pragma unroll
                for (int r = 0; r < 8; ++r) {
                    const int voxel = base + r + 8 * khalf;
                    if (voxel < n) {
                        float g = 0.f;
#pragma unroll
                        for (int c = 0; c < 3; ++c) {
                            const float bnv = acc[c][cc][r] * scale[c] + shift[c];
                            g += 1.f / (1.f + __expf(-bnv));
                        }
                        op[r * C_OUT] = g * yp[r * C_OUT];
                    }
                }
            }
        }
    }
}

extern "C" void kernel_launch(void* const* d_in, const int* in_sizes, int n_in,
                              void* d_out, int out_size, void* d_ws, size_t ws_size,
                              hipStream_t stream)
{
    const float* x    = (const float*)d_in[0];
    const int*   nbr  = (const int*)  d_in[1];
    const float* W1   = (const float*)d_in[2];
    const float* W12  = (const float*)d_in[3];
    const float* W2   = (const float*)d_in[4];
    const float* W3   = (const float*)d_in[5];
    const float* Wr1  = (const float*)d_in[6];
    const float* Wr2  = (const float*)d_in[7];
    const float* Wr3  = (const float*)d_in[8];
    const float* bnp  = (const float*)d_in[9];   // [7,4,32]

    const int n = in_sizes[0] / 16;              // x_feat is [n,16]

    float* t1 = (float*)d_ws;                    // conv1 out / r-intermediate
    float* t2 = t1 + (size_t)n * C_OUT;          // s
    float* t3 = t2 + (size_t)n * C_OUT;          // y

    const Offs9 p133 = {{ 9, 10, 11, 12, 13, 14, 15, 16, 17}};
    const Offs9 p313 = {{ 3,  4,  5, 12, 13, 14, 21, 22, 23}};

    const int ntiles = (n + 15) / 16;
    const dim3 blk(256);                 // 8 wave32s per block
    const int  grid = (ntiles + 7) / 8;  // one 16-voxel tile per wave

    // ResContextBlock
    subm_conv_wmma<16, 9><<<grid, blk, 0, stream>>>(x,  nbr, W1,  bnp + 0 * 128, nullptr, t1, n, p133, MODE_LRELU_BN);
    subm_conv_wmma<32, 9><<<grid, blk, 0, stream>>>(t1, nbr, W12, bnp + 1 * 128, nullptr, t2, n, p313, MODE_LRELU_BN);
    subm_conv_wmma<16, 9><<<grid, blk, 0, stream>>>(x,  nbr, W2,  bnp + 2 * 128, nullptr, t1, n, p313, MODE_LRELU_BN);
    subm_conv_wmma<32, 9><<<grid, blk, 0, stream>>>(t1, nbr, W3,  bnp + 3 * 128, t2,      t3, n, p133, MODE_LRELU_BN_ADD); // y = r + s
    // Fused ReconBlock + gating multiply, straight to d_out
    recon_gate_wmma<<<grid, blk, 0, stream>>>(t3, nbr, Wr1, Wr2, Wr3, bnp + 4 * 128, (float*)d_out, n);
}